// ADesigner_8040178778146
// MI455X (gfx1250) — compile-verified
//
#include <hip/hip_runtime.h>

typedef float v2f __attribute__((ext_vector_type(2)));
typedef float v8f __attribute__((ext_vector_type(8)));

#define TOK_BOA 21
#define TOK_BOH 22
#define TOK_BOL 23

// ---------------------------------------------------------------------------
// Kernel 1: faithful "last global token" scan -> per-node segment id + flag.
// seg[i] = gval of most recent global token at index <= i, falling back to
// gval[0] (reproduces reference's gval[clip(last_glb, 0)]).
// meta[i] = seg | (is_global << 8)
// ---------------------------------------------------------------------------
__global__ __launch_bounds__(1024) void seg_meta_kernel(
    const int* __restrict__ S, int n, int* __restrict__ meta)
{
    __shared__ int sval[1024];
    __shared__ int carryS;
    const int tid = threadIdx.x;

    if (tid == 0) {
        const int s0 = S[0];
        carryS = (s0 == TOK_BOA) ? 1 : (s0 == TOK_BOH) ? 2 : (s0 == TOK_BOL) ? 3 : 0;
    }

    for (int base = 0; base < n; base += 1024) {
        const int i = base + tid;
        int g = 0, isg = 0;
        if (i < n) {
            const int s = S[i];
            if      (s == TOK_BOA) { g = 1; isg = 1; }
            else if (s == TOK_BOH) { g = 2; isg = 1; }
            else if (s == TOK_BOL) { g = 3; isg = 1; }
        }
        sval[tid] = g;
        __syncthreads();
        // Hillis-Steele inclusive scan, operator = "rightmost non-zero"
        for (int off = 1; off < 1024; off <<= 1) {
            const int prev = (tid >= off) ? sval[tid - off] : 0;
            const int mine = sval[tid];
            __syncthreads();
            if (mine == 0 && prev != 0) sval[tid] = prev;
            __syncthreads();
        }
        int segv = sval[tid];
        if (segv == 0) segv = carryS;           // carry from previous chunks
        if (i < n) meta[i] = segv | (isg << 8);
        __syncthreads();
        if (tid == 0) {
            const int last = sval[1023];
            if (last != 0) carryS = last;
        }
        __syncthreads();
    }
}

// ---------------------------------------------------------------------------
// Kernel 2: one wave32 per 16x16 (row, col) pair tile.
//   d2 = (norm_r - 2 * x_r . x_c) + norm_c
// computed by one V_WMMA_F32_16X16X4_F32:
//   A (16x4): K=0..2 = row coords, K=3 = row norm
//   B (4x16): K=0..2 = -2 * col coords, K=3 = 1.0
// C/D layout: VGPR j, lane L -> row = row0 + j + 8*(L>=16), col = col0 + L%16
// Epilogue is fully branchless; row metadata is fetched once per wave and
// redistributed with lane shuffles.
// ---------------------------------------------------------------------------
__global__ __launch_bounds__(256) void edge_pair_kernel(
    const float* __restrict__ X, const int* __restrict__ offsets,
    const int* __restrict__ meta, float* __restrict__ out,
    int n, int nb, int maxn)
{
    const int lane      = threadIdx.x & 31;
    const int waveInBlk = threadIdx.x >> 5;
    const int colTiles  = maxn >> 4;
    const int tile      = blockIdx.x * (blockDim.x >> 5) + waveInBlk;
    const int rowTile   = tile / colTiles;
    const int colTile   = tile - rowTile * colTiles;
    const int row0      = rowTile << 4;
    if (row0 >= n) return;                 // wave-uniform: EXEC stays all-ones
    const int col0 = colTile << 4;

    // Batch lookup (wave-uniform scalar loop; rows of a tile share a batch)
    int b = 0;
    while (b + 1 < nb && offsets[b + 1] <= row0) ++b;
    const int node_off = offsets[b];
    const int len      = offsets[b + 1] - node_off;

    const int half = lane >> 4;            // 0: lanes 0-15, 1: lanes 16-31
    const int l15  = lane & 15;

    // ---- A operand: row coordinates (channel 1 of X[N,4,3]) + row norm ----
    const int    r  = row0 + l15;
    const float* xr = X + (size_t)r * 12 + 3;
    const float rx = xr[0], ry = xr[1], rz = xr[2];
    const float rnorm = rx * rx + ry * ry + rz * rz;
    v2f A;
    A.x = half ? rz    : rx;               // K=0 (lo) / K=2 (hi)
    A.y = half ? rnorm : ry;               // K=1 (lo) / K=3 (hi)

    // ---- B operand: -2 * column coordinates, ones at K=3 ----
    const int c  = col0 + l15;
    int cg = c + node_off;                 // reference clips col_g to n-1
    if (cg > n - 1) cg = n - 1;
    const float* xc = X + (size_t)cg * 12 + 3;
    const float cx = xc[0], cy = xc[1], cz = xc[2];
    const float cnorm = cx * cx + cy * cy + cz * cz;
    v2f Bm;
    Bm.x = half ? (-2.0f * cz) : (-2.0f * cx);
    Bm.y = half ? 1.0f         : (-2.0f * cy);

    // Row metadata for the whole tile, one load per lane, shuffled later.
    const int rmetaLane = meta[row0 + l15];      // lane l (l<16) -> row0 + l

    v8f acc = {};
    acc = __builtin_amdgcn_wmma_f32_16x16x4_f32(false, A, false, Bm,
                                                (short)0, acc, false, false);
    // acc[j] = rnorm(row_j) - 2 * dot(row_j, col)

    // ---- per-lane column metadata (constant over j) ----
    const int cmeta = meta[cg];
    const int cseg  = cmeta & 0xff;
    const int cglb  = (cmeta >> 8) & 1;
    const int colOk = (int)(c < len);

    const size_t planeStride = (size_t)n * (size_t)maxn;
    const int    rowBase     = row0 + (half << 3);
    const int    rlBase      = rowBase - node_off;
    const int    didxBase    = rowBase - cg;

#pragma unroll
    for (int j = 0; j < 8; ++j) {
        const int rmeta = __shfl(rmetaLane, j + (half << 3), 32);
        const int rseg  = rmeta & 0xff;
        const int rglb  = (rmeta >> 8) & 1;

        const float d2    = acc[j] + cnorm;
        const int   valid = colOk & (int)(c != (rlBase + j));
        const int   anyg  = rglb | cglb;
        const int   notg  = anyg ^ 1;
        const int   same  = (int)(rseg == cseg);
        const int   didx  = didxBase + j;
        const int   seq   = ((int)(didx == 1) | (int)(didx == -1))
                          & notg & (int)(rseg != 1);

        const int ctx = valid & ( (notg & same & (int)(d2 <= 64.0f))  // ctx_radial
                                | (same & anyg)                       // global_normal
                                | (rglb & cglb)                       // global_global
                                | seq );                              // seq_adj
        const int inter = valid & notg & (same ^ 1) & (int)(d2 <= 144.0f);

        const size_t o = (size_t)(rowBase + j) * (size_t)maxn + (size_t)c;
        out[o]               = ctx   ? 1.0f : 0.0f;
        out[planeStride + o] = inter ? 1.0f : 0.0f;
    }
}

// ---------------------------------------------------------------------------
extern "C" void kernel_launch(void* const* d_in, const int* in_sizes, int n_in,
                              void* d_out, int out_size, void* d_ws, size_t ws_size,
                              hipStream_t stream)
{
    const float* X       = (const float*)d_in[0];   // [N, 4, 3] f32
    const int*   S       = (const int*)d_in[1];     // [N] i32
    const int*   offsets = (const int*)d_in[2];     // [B+1] i32
    float*       out     = (float*)d_out;           // [2, N, MAX_N] f32

    const int n    = in_sizes[0] / 12;
    const int nb   = in_sizes[2] - 1;
    const int maxn = out_size / (2 * n);

    int* meta = (int*)d_ws;                         // n * 4 bytes of scratch

    seg_meta_kernel<<<1, 1024, 0, stream>>>(S, n, meta);

    const int tiles       = (n >> 4) * (maxn >> 4); // 16x16 pair tiles
    const int wavesPerBlk = 8;                      // 256 threads, wave32
    const int blocks      = (tiles + wavesPerBlk - 1) / wavesPerBlk;
    edge_pair_kernel<<<blocks, 256, 0, stream>>>(X, offsets, meta, out, n, nb, maxn);
}